// GCN_73383811219606
// MI455X (gfx1250) — compile-verified
//
#include <hip/hip_runtime.h>

typedef __attribute__((ext_vector_type(2))) float v2f;
typedef __attribute__((ext_vector_type(8))) float v8f;

#define GCN_N 50000
#define GCN_E 800000

// ---------------- degree / normalization ----------------

__global__ void zero_f32_kernel(float* __restrict__ p, int n) {
    int i = blockIdx.x * blockDim.x + threadIdx.x;
    if (i < n) p[i] = 0.0f;
}

__global__ void degree_kernel(const long long* __restrict__ ei, float* __restrict__ deg, int E) {
    int e = blockIdx.x * blockDim.x + threadIdx.x;
    if (e < E) {
        long long d = ei[E + e];
        atomicAdd(&deg[(int)d], 1.0f);
    }
}

// in-place: deg -> rsqrt(deg + 1)
__global__ void rsqrt_kernel(float* __restrict__ p, int n) {
    int i = blockIdx.x * blockDim.x + threadIdx.x;
    if (i < n) p[i] = rsqrtf(p[i] + 1.0f);
}

// ---------------- fp32 WMMA GEMM: C[M,N] = (relu?)A[M,K] @ B[K,N] ----------------
// One wave -> ROWS x 64 tile (ROWS = 32, remainder tile = 16).
// N, K compile-time: all inner-loop addressing strength-reduces to ptr+imm.

template <bool RELU_A, int N, int K, int ROWS>
__device__ __forceinline__ void gemm_tile(const float* __restrict__ A,
                                          const float* __restrict__ B,
                                          float* __restrict__ C,
                                          int row0, int ns, int lane)
{
    const int half = lane >> 4;   // 0: lanes 0-15, 1: lanes 16-31
    const int l    = lane & 15;
    const int col0 = ns * 64 + l;
    const int RT   = ROWS / 16;   // 1 or 2 row-tiles

    // A 16x4 frag: VGPR0 = K=k+2*half, VGPR1 = K=k+1+2*half (8B-aligned pair)
    const float* Arow = A + (size_t)(row0 + l) * K + 2 * half;
    // B 4x16 frag: VGPR0 = row k+2*half, VGPR1 = row k+1+2*half, N across lanes
    const float* Brow = B + (size_t)(2 * half) * N + col0;

    v8f acc[RT][4];
#pragma unroll
    for (int r = 0; r < RT; ++r)
#pragma unroll
        for (int j = 0; j < 4; ++j) acc[r][j] = (v8f){};

    for (int k = 0; k < K; k += 4) {
        v2f a[RT];
#pragma unroll
        for (int r = 0; r < RT; ++r) {
            a[r] = *(const v2f*)(Arow + (size_t)r * 16 * K + k);
            if (RELU_A) { a[r].x = fmaxf(a[r].x, 0.0f); a[r].y = fmaxf(a[r].y, 0.0f); }
        }
        const float* bp = Brow + (size_t)k * N;
        v2f b[4];
#pragma unroll
        for (int j = 0; j < 4; ++j) {
            b[j].x = bp[16 * j];        // row k+2*half,   imm offset 16j*4
            b[j].y = bp[N + 16 * j];    // row k+1+2*half, imm offset (N+16j)*4
        }
#pragma unroll
        for (int r = 0; r < RT; ++r) {
#pragma unroll
            for (int j = 0; j < 4; ++j)
                acc[r][j] = __builtin_amdgcn_wmma_f32_16x16x4_f32(
                    false, a[r], false, b[j], (short)0, acc[r][j], false, false);
        }
    }

    // C/D 16x16: VGPR i -> row M = i + 8*half, col N = l
#pragma unroll
    for (int r = 0; r < RT; ++r) {
        float* Cbase = C + (size_t)(row0 + r * 16 + 8 * half) * N + ns * 64 + l;
#pragma unroll
        for (int i = 0; i < 8; ++i) {
            float* cp = Cbase + (size_t)i * N;
            cp[0]  = acc[r][0][i];
            cp[16] = acc[r][1][i];
            cp[32] = acc[r][2][i];
            cp[48] = acc[r][3][i];
        }
    }
}

template <bool RELU_A, int N, int K>
__global__ __launch_bounds__(256) void gemm_wmma_kernel(
    const float* __restrict__ A, const float* __restrict__ B,
    float* __restrict__ C, int M)
{
    const int lane    = threadIdx.x & 31;
    const int nstrips = N / 64;
    const int mtiles  = (M + 31) >> 5;                 // 32-row supertiles
    const int wid     = blockIdx.x * 8 + (threadIdx.x >> 5);
    if (wid >= mtiles * nstrips) return;               // wave-uniform guard
    const int mt   = wid / nstrips;
    const int ns   = wid % nstrips;
    const int row0 = mt * 32;
    if (row0 + 32 <= M)                                // wave-uniform branch
        gemm_tile<RELU_A, N, K, 32>(A, B, C, row0, ns, lane);
    else
        gemm_tile<RELU_A, N, K, 16>(A, B, C, row0, ns, lane);
}

// ---------------- aggregation ----------------

// out[i,f] = h[i,f] * dis[i]^2 + bias[f]   (self-loop term + bias, init accumulator)
template <int F>
__global__ void init_out_kernel(float* __restrict__ out, const float* __restrict__ h,
                                const float* __restrict__ dis, const float* __restrict__ bias,
                                int n)
{
    long long idx = (long long)blockIdx.x * blockDim.x + threadIdx.x;
    if (idx >= (long long)n * F) return;
    int i = (int)(idx / F);
    int f = (int)(idx % F);
    float di = dis[i];
    out[idx] = h[idx] * di * di + bias[f];
}

// out[dst,f] += h[src,f] * dis[src]*dis[dst]
template <int LOG2F>
__global__ void scatter_kernel(float* __restrict__ out, const float* __restrict__ h,
                               const float* __restrict__ dis,
                               const long long* __restrict__ ei, int E)
{
    const int F = 1 << LOG2F;
    long long idx = (long long)blockIdx.x * blockDim.x + threadIdx.x;
    if (idx >= ((long long)E << LOG2F)) return;
    int e = (int)(idx >> LOG2F);
    int f = (int)(idx & (F - 1));
    int s = (int)ei[e];
    int d = (int)ei[E + e];
    float nrm = dis[s] * dis[d];
    atomicAdd(&out[(size_t)d * F + f], h[(size_t)s * F + f] * nrm);
}

// ---------------- classifier: logits = relu(out2) @ Wc + bc ; log_softmax ----------------

__global__ void classifier_kernel(const float* __restrict__ out2, const float* __restrict__ Wc,
                                  const float* __restrict__ bc, float* __restrict__ out, int n)
{
    int i = blockIdx.x * blockDim.x + threadIdx.x;
    if (i >= n) return;
    float l0 = bc[0], l1 = bc[1];
    const float* row = out2 + (size_t)i * 128;
#pragma unroll 4
    for (int f = 0; f < 128; ++f) {
        float v = fmaxf(row[f], 0.0f);
        l0 += v * Wc[2 * f + 0];
        l1 += v * Wc[2 * f + 1];
    }
    float m   = fmaxf(l0, l1);
    float lse = m + logf(expf(l0 - m) + expf(l1 - m));
    out[2 * i + 0] = l0 - lse;
    out[2 * i + 1] = l1 - lse;
}

// ---------------- driver ----------------

extern "C" void kernel_launch(void* const* d_in, const int* in_sizes, int n_in,
                              void* d_out, int out_size, void* d_ws, size_t ws_size,
                              hipStream_t stream)
{
    const float*     x  = (const float*)d_in[0];
    const long long* ei = (const long long*)d_in[1];
    const float*     W1 = (const float*)d_in[2];
    const float*     b1 = (const float*)d_in[3];
    const float*     W2 = (const float*)d_in[4];
    const float*     b2 = (const float*)d_in[5];
    const float*     Wc = (const float*)d_in[6];
    const float*     bc = (const float*)d_in[7];
    float* out = (float*)d_out;

    const int N = GCN_N, E = GCN_E;

    // workspace layout (floats)
    float* dis  = (float*)d_ws;                 //   50,000  (padded to 51,200)
    float* h1   = dis  + 51200;                 // 25,600,000
    float* out1 = h1   + (size_t)N * 512;       // 25,600,000
    float* h2   = out1 + (size_t)N * 512;       //  6,400,000
    float* out2 = h2   + (size_t)N * 128;       //  6,400,000

    // 1) symmetric normalization:  dis = rsqrt(in_degree + 1)
    zero_f32_kernel<<<(N + 255) / 256, 256, 0, stream>>>(dis, N);
    degree_kernel<<<(E + 255) / 256, 256, 0, stream>>>(ei, dis, E);
    rsqrt_kernel<<<(N + 255) / 256, 256, 0, stream>>>(dis, N);

    // 2) h1 = x @ W1          (50000x768 @ 768x512)
    {
        int waves = ((N + 31) / 32) * (512 / 64);   // 1563 * 8 = 12504
        gemm_wmma_kernel<false, 512, 768><<<(waves + 7) / 8, 256, 0, stream>>>(x, W1, h1, N);
    }
    // 3) out1 = scatter(h1*norm) + h1*dis^2 + b1
    init_out_kernel<512><<<(int)(((long long)N * 512 + 255) / 256), 256, 0, stream>>>(out1, h1, dis, b1, N);
    scatter_kernel<9><<<(int)(((long long)E * 512 + 255) / 256), 256, 0, stream>>>(out1, h1, dis, ei, E);

    // 4) h2 = relu(out1) @ W2 (50000x512 @ 512x128), relu folded into A-load
    {
        int waves = ((N + 31) / 32) * (128 / 64);   // 1563 * 2 = 3126
        gemm_wmma_kernel<true, 128, 512><<<(waves + 7) / 8, 256, 0, stream>>>(out1, W2, h2, N);
    }
    // 5) out2 = scatter(h2*norm) + h2*dis^2 + b2
    init_out_kernel<128><<<(int)(((long long)N * 128 + 255) / 256), 256, 0, stream>>>(out2, h2, dis, b2, N);
    scatter_kernel<7><<<(int)(((long long)E * 128 + 255) / 256), 256, 0, stream>>>(out2, h2, dis, ei, E);

    // 6) logits + log_softmax (relu folded in)
    classifier_kernel<<<(N + 255) / 256, 256, 0, stream>>>(out2, Wc, bc, out, N);
}